// GeneralConv_24421184045728
// MI455X (gfx1250) — compile-verified
//
#include <hip/hip_runtime.h>
#include <hip/hip_bf16.h>
#include <math.h>

// GeneralConv (hetero GAT-style) for MI455X / gfx1250, wave32.
//
// Algebraic restructure (exact, uses linearity of Wk/Wv):
//   aq[n,h]   = (x_n+emb) . (att_i^T Wq)[h]        (per node,   [N,4])
//   ak[n,h]   = (x_n+emb) . (att_j^T Wk)[h]        (per node,   [N,4])
//   ak_e[t,h] = edge_emb[t] . (att_j^T Wk)[h]      (per e-type, [8,4])
//   xv_node   = (x+emb) @ Wv^T                     (WMMA f32 GEMM, [N,128])
//   xv_e      = edge_emb @ Wv^T                    ([8,128])
//   alpha_e   = leaky(aq[dst] + ak[src] + ak_e[et])
//   out[dst] += exp(alpha_e - amax[dst]) * (xv_node[src] + xv_e[et])
//   out[n]    = out[n] / (denom[n] + 1e-16) + bias
// This removes the 680k-edge 128x128 GEMMs entirely (35x FLOP cut); the run
// is then bound by L2-resident gather/atomic traffic (~0.75 GB @ L2).

typedef __attribute__((ext_vector_type(2))) float v2f;
typedef __attribute__((ext_vector_type(4))) float v4f;
typedef __attribute__((ext_vector_type(8))) float v8f;

#define DD 128
#define HH 4
#define CC 32
#define HC 128
#define NEG_SLOPE 0.2f
#define NUM_ET 8

// monotonic uint encoding of float for atomicMax-based segment max
__device__ __forceinline__ unsigned fkey(float f) {
  unsigned b = __float_as_uint(f);
  return (b & 0x80000000u) ? ~b : (b | 0x80000000u);
}
__device__ __forceinline__ float funkey(unsigned k) {
  unsigned b = (k & 0x80000000u) ? (k & 0x7fffffffu) : ~k;
  return __uint_as_float(b);
}

// ---------------------------------------------------------------------------
// Tiny weight-folding kernel (1 block, 128 threads)
__global__ void prep_weights(const float* __restrict__ Wq, const float* __restrict__ Wk,
                             const float* __restrict__ Wv,
                             const float* __restrict__ att_i, const float* __restrict__ att_j,
                             const float* __restrict__ edge_emb,
                             float* __restrict__ wq_att, float* __restrict__ wk_att,
                             float* __restrict__ xv_edge, float* __restrict__ ak_edge) {
  const int t = threadIdx.x;  // 0..127
#pragma unroll
  for (int h = 0; h < HH; ++h) {
    float sq = 0.f, sk = 0.f;
    for (int c = 0; c < CC; ++c) {
      sq += att_i[h * CC + c] * Wq[(h * CC + c) * DD + t];
      sk += att_j[h * CC + c] * Wk[(h * CC + c) * DD + t];
    }
    wq_att[h * DD + t] = sq;
    wk_att[h * DD + t] = sk;
  }
  for (int et = 0; et < NUM_ET; ++et) {
    float s = 0.f;
    for (int d = 0; d < DD; ++d) s += edge_emb[et * DD + d] * Wv[t * DD + d];
    xv_edge[et * HC + t] = s;
  }
  __syncthreads();
  if (t < NUM_ET * HH) {
    const int et = t >> 2, h = t & 3;
    float s = 0.f;
    for (int d = 0; d < DD; ++d) s += edge_emb[et * DD + d] * wk_att[h * DD + d];
    ak_edge[et * HH + h] = s;
  }
}

// ---------------------------------------------------------------------------
// Node phase: x += node_emb[type]; aq/ak scalars; zero-init accumulators.
// One wave per node, lane l owns channels 4l..4l+3.
__global__ void node_phase(const float* __restrict__ x, const int* __restrict__ node_type,
                           const float* __restrict__ node_emb,
                           const float* __restrict__ wq_att, const float* __restrict__ wk_att,
                           float* __restrict__ xnode, float* __restrict__ aq,
                           float* __restrict__ ak, unsigned* __restrict__ amax_key,
                           float* __restrict__ denom, float* __restrict__ out_accum, int N) {
  const int wave = threadIdx.x >> 5;
  const int lane = threadIdx.x & 31;
  const int n = blockIdx.x * 8 + wave;
  if (n >= N) return;
  const int nt = node_type[n];
  v4f xv = *(const v4f*)(x + (size_t)n * DD + lane * 4);
  const v4f ev = *(const v4f*)(node_emb + (size_t)nt * DD + lane * 4);
  xv += ev;
  *(v4f*)(xnode + (size_t)n * DD + lane * 4) = xv;

  float pq[HH], pk[HH];
#pragma unroll
  for (int h = 0; h < HH; ++h) {
    const v4f wq4 = *(const v4f*)(wq_att + h * DD + lane * 4);
    const v4f wk4 = *(const v4f*)(wk_att + h * DD + lane * 4);
    pq[h] = xv.x * wq4.x + xv.y * wq4.y + xv.z * wq4.z + xv.w * wq4.w;
    pk[h] = xv.x * wk4.x + xv.y * wk4.y + xv.z * wk4.z + xv.w * wk4.w;
  }
#pragma unroll
  for (int h = 0; h < HH; ++h)
    for (int off = 16; off > 0; off >>= 1) {
      pq[h] += __shfl_down(pq[h], off, 32);
      pk[h] += __shfl_down(pk[h], off, 32);
    }
  if (lane == 0) {
#pragma unroll
    for (int h = 0; h < HH; ++h) {
      aq[(size_t)n * HH + h] = pq[h];
      ak[(size_t)n * HH + h] = pk[h];
    }
  }
  // zero-init accumulators (ws and d_out are poisoned by the harness)
  *(v4f*)(out_accum + (size_t)n * HC + lane * 4) = (v4f){0.f, 0.f, 0.f, 0.f};
  if (lane < HH) {
    amax_key[(size_t)n * HH + lane] = 0u;  // below any real key
    denom[(size_t)n * HH + lane] = 0.f;
  }
}

// ---------------------------------------------------------------------------
// xv_node[N,128] = xnode[N,128] @ Wv^T using V_WMMA_F32_16X16X4_F32.
// Block = 8 waves; wave w owns output cols [16w,16w+16); block owns 16 rows.
// A layout (16x4 f32): lane 0-15 -> M=lane, K={0,1}; lane 16-31 -> M=lane-16, K={2,3}.
// B layout mirrors A: B[k][n] with n=lane&15, k offset klo. B[k][n] = Wv[n][k],
// so b-pairs are contiguous Wv row elements -> plain b64 loads, preloaded to VGPRs.
__global__ void __launch_bounds__(256) gemm_xv(const float* __restrict__ xnode,
                                               const float* __restrict__ Wv,
                                               float* __restrict__ xv_node, int N) {
  const int wave = threadIdx.x >> 5;
  const int lane = threadIdx.x & 31;
  const int colbase = wave * 16;
  const int nodebase = blockIdx.x * 16;
  const int m = lane & 15;
  const int klo = (lane >> 4) * 2;

  v2f breg[32];
#pragma unroll
  for (int kk = 0; kk < 32; ++kk)
    breg[kk] = *(const v2f*)(Wv + (size_t)(colbase + m) * DD + kk * 4 + klo);

  int row = nodebase + m;
  if (row >= N) row = N - 1;  // clamp reads; keeps EXEC all-ones for WMMA
  const float* aptr = xnode + (size_t)row * DD + klo;

  v8f acc = {};
#pragma unroll
  for (int kk = 0; kk < 32; ++kk) {
    const v2f a = *(const v2f*)(aptr + kk * 4);
    acc = __builtin_amdgcn_wmma_f32_16x16x4_f32(false, a, false, breg[kk],
                                                (short)0, acc, false, false);
  }
  // D layout: VGPR r -> lanes 0-15: (M=r, N=lane); lanes 16-31: (M=r+8, N=lane-16)
#pragma unroll
  for (int r = 0; r < 8; ++r) {
    const int orow = nodebase + ((lane < 16) ? r : (r + 8));
    if (orow < N) xv_node[(size_t)orow * DD + colbase + m] = acc[r];
  }
}

// ---------------------------------------------------------------------------
// Edge pass 1: segment max of leaky-relu attention logits (atomicMax on key).
__global__ void edge_max(const int* __restrict__ src, const int* __restrict__ dst,
                         const int* __restrict__ etype, const float* __restrict__ aq,
                         const float* __restrict__ ak, const float* __restrict__ ak_edge,
                         unsigned* __restrict__ amax_key, int E, int total) {
  const int e = blockIdx.x * blockDim.x + threadIdx.x;
  if (e >= total) return;
  int s, d, t;
  if (e < E) { s = src[e]; d = dst[e]; t = etype[e]; }
  else       { s = d = e - E; t = 0; }  // self-loop, EDGE_TYPE_SELF=0
  const v4f a = *(const v4f*)(aq + (size_t)d * HH);
  const v4f b = *(const v4f*)(ak + (size_t)s * HH);
  const v4f c = *(const v4f*)(ak_edge + (size_t)t * HH);
#pragma unroll
  for (int h = 0; h < HH; ++h) {
    float al = a[h] + b[h] + c[h];
    al = (al < 0.f) ? al * NEG_SLOPE : al;
    atomicMax(amax_key + (size_t)d * HH + h, fkey(al));
  }
}

// ---------------------------------------------------------------------------
// Edge pass 2: w = exp(alpha - amax[dst]); denom[dst]+=w;
// out[dst] += w * (xv_node[src] + xv_edge[et]).   One wave per edge.
__global__ void edge_accum(const int* __restrict__ src, const int* __restrict__ dst,
                           const int* __restrict__ etype, const float* __restrict__ aq,
                           const float* __restrict__ ak, const float* __restrict__ ak_edge,
                           const unsigned* __restrict__ amax_key, float* __restrict__ denom,
                           const float* __restrict__ xv_node, const float* __restrict__ xv_edge,
                           float* __restrict__ out_accum, int E, int total) {
  const int wave = threadIdx.x >> 5;
  const int lane = threadIdx.x & 31;
  const int e = blockIdx.x * 8 + wave;
  if (e >= total) return;
  int s, d, t;
  if (e < E) { s = src[e]; d = dst[e]; t = etype[e]; }
  else       { s = d = e - E; t = 0; }
  const v4f a = *(const v4f*)(aq + (size_t)d * HH);
  const v4f b = *(const v4f*)(ak + (size_t)s * HH);
  const v4f c = *(const v4f*)(ak_edge + (size_t)t * HH);
  float w0, w1, w2, w3;
  {
    float al0 = a[0] + b[0] + c[0]; al0 = (al0 < 0.f) ? al0 * NEG_SLOPE : al0;
    float al1 = a[1] + b[1] + c[1]; al1 = (al1 < 0.f) ? al1 * NEG_SLOPE : al1;
    float al2 = a[2] + b[2] + c[2]; al2 = (al2 < 0.f) ? al2 * NEG_SLOPE : al2;
    float al3 = a[3] + b[3] + c[3]; al3 = (al3 < 0.f) ? al3 * NEG_SLOPE : al3;
    w0 = __expf(al0 - funkey(amax_key[(size_t)d * HH + 0]));
    w1 = __expf(al1 - funkey(amax_key[(size_t)d * HH + 1]));
    w2 = __expf(al2 - funkey(amax_key[(size_t)d * HH + 2]));
    w3 = __expf(al3 - funkey(amax_key[(size_t)d * HH + 3]));
  }
  if (lane < HH) {
    const float wl = (lane == 0) ? w0 : (lane == 1) ? w1 : (lane == 2) ? w2 : w3;
    atomicAdd(denom + (size_t)d * HH + lane, wl);
  }
  const int c0 = lane * 4;          // 4 channels per lane, one head per 8 lanes
  const int h = lane >> 3;
  const float wh = (h == 0) ? w0 : (h == 1) ? w1 : (h == 2) ? w2 : w3;
  const v4f vn = *(const v4f*)(xv_node + (size_t)s * HC + c0);
  const v4f ve = *(const v4f*)(xv_edge + (size_t)t * HC + c0);
  float* op = out_accum + (size_t)d * HC + c0;
  atomicAdd(op + 0, wh * (vn.x + ve.x));
  atomicAdd(op + 1, wh * (vn.y + ve.y));
  atomicAdd(op + 2, wh * (vn.z + ve.z));
  atomicAdd(op + 3, wh * (vn.w + ve.w));
}

// ---------------------------------------------------------------------------
__global__ void finalize(float* __restrict__ out, const float* __restrict__ denom,
                         const float* __restrict__ bias, int N) {
  const int idx = blockIdx.x * blockDim.x + threadIdx.x;
  if (idx >= N * HC) return;
  const int n = idx >> 7, ch = idx & 127, h = ch >> 5;
  out[idx] = out[idx] / (denom[(size_t)n * HH + h] + 1e-16f) + bias[ch];
}

// ---------------------------------------------------------------------------
extern "C" void kernel_launch(void* const* d_in, const int* in_sizes, int n_in,
                              void* d_out, int out_size, void* d_ws, size_t ws_size,
                              hipStream_t stream) {
  (void)n_in; (void)out_size; (void)ws_size;
  const float* x          = (const float*)d_in[0];
  const int*   edge_index = (const int*)d_in[1];
  const int*   node_type  = (const int*)d_in[2];
  const int*   edge_type  = (const int*)d_in[3];
  const float* Wq         = (const float*)d_in[4];
  const float* Wk         = (const float*)d_in[5];
  const float* Wv         = (const float*)d_in[6];
  const float* att_i      = (const float*)d_in[7];
  const float* att_j      = (const float*)d_in[8];
  const float* bias       = (const float*)d_in[9];
  const float* node_emb   = (const float*)d_in[10];
  const float* edge_emb   = (const float*)d_in[11];

  const int N = in_sizes[0] / DD;
  const int E = in_sizes[3];
  const int* src = edge_index;
  const int* dst = edge_index + E;
  const int total = E + N;

  float* ws = (float*)d_ws;
  size_t off = 0;
  float*    xnode    = ws + off; off += (size_t)N * DD;
  float*    xv_node  = ws + off; off += (size_t)N * DD;
  float*    aq       = ws + off; off += (size_t)N * HH;
  float*    ak       = ws + off; off += (size_t)N * HH;
  unsigned* amax_key = (unsigned*)(ws + off); off += (size_t)N * HH;
  float*    denom    = ws + off; off += (size_t)N * HH;
  float*    wq_att   = ws + off; off += HH * DD;
  float*    wk_att   = ws + off; off += HH * DD;
  float*    xv_edge  = ws + off; off += NUM_ET * HC;
  float*    ak_edge  = ws + off; off += NUM_ET * HH;
  float*    out      = (float*)d_out;

  prep_weights<<<1, 128, 0, stream>>>(Wq, Wk, Wv, att_i, att_j, edge_emb,
                                      wq_att, wk_att, xv_edge, ak_edge);
  node_phase<<<(N + 7) / 8, 256, 0, stream>>>(x, node_type, node_emb, wq_att, wk_att,
                                              xnode, aq, ak, amax_key, denom, out, N);
  gemm_xv<<<(N + 15) / 16, 256, 0, stream>>>(xnode, Wv, xv_node, N);
  edge_max<<<(total + 255) / 256, 256, 0, stream>>>(src, dst, edge_type, aq, ak, ak_edge,
                                                    amax_key, E, total);
  edge_accum<<<(total + 7) / 8, 256, 0, stream>>>(src, dst, edge_type, aq, ak, ak_edge,
                                                  amax_key, denom, xv_node, xv_edge,
                                                  out, E, total);
  finalize<<<((N * HC) + 255) / 256, 256, 0, stream>>>(out, denom, bias, N);
}